// HGCNSheafBuilderDiag_28260884808002
// MI455X (gfx1250) — compile-verified
//
#include <hip/hip_runtime.h>
#include <math.h>

#define NNZ_C   1600000
#define NNODE   50000
#define NEDGE   25000
#define DD      6
#define FF      64
#define TWOF    128
#define PADW    132     // 128 + 4 pad: spreads rows across LDS banks
#define WPB     4       // waves (= 16-row tiles) per block

typedef float v2f __attribute__((ext_vector_type(2)));
typedef float v8f __attribute__((ext_vector_type(8)));

// ---------------------------------------------------------------------------
// Kernel 0: fold gamma into W, fold beta@W^T into bias. Tiny, one block.
// gWg layout: 16 rows x 128 (rows >= 6 are zero) ; gBB: 16 floats
// ---------------------------------------------------------------------------
__global__ void prep_w_kernel(const float* __restrict__ W,
                              const float* __restrict__ gamma,
                              const float* __restrict__ beta,
                              const float* __restrict__ b,
                              float* __restrict__ gWg,
                              float* __restrict__ gBB) {
  int t = threadIdx.x;
  for (int i = t; i < 16 * TWOF; i += blockDim.x) {
    int n = i >> 7, k = i & 127;
    gWg[i] = (n < DD) ? W[n * TWOF + k] * gamma[k] : 0.0f;
  }
  if (t < 16) {
    float acc = (t < DD) ? b[t] : 0.0f;
    if (t < DD) {
      for (int k = 0; k < TWOF; ++k) acc += beta[k] * W[t * TWOF + k];
    }
    gBB[t] = acc;
  }
}

// ---------------------------------------------------------------------------
// Kernel 1/2: collapse D=6 stalk copies by mean.  src: (rows*6, 64) -> dst (rows,64)
// 64 consecutive threads read 6 contiguous 256B rows -> fully coalesced.
// ---------------------------------------------------------------------------
__global__ void mean6_kernel(const float* __restrict__ src,
                             float* __restrict__ dst, int total) {
  int i = blockIdx.x * blockDim.x + threadIdx.x;
  if (i >= total) return;
  int n = i >> 6, f = i & 63;
  const float* p = src + (size_t)n * (DD * FF) + f;
  float s = p[0] + p[64] + p[128] + p[192] + p[256] + p[320];
  dst[i] = s * (1.0f / 6.0f);
}

// ---------------------------------------------------------------------------
// Main kernel: one wave = one 16-nnz tile.
//   gather 16x128 f32 -> LDS, LayerNorm stats per row (2 lanes/row + SWAPX16),
//   32 x V_WMMA_F32_16X16X4_F32 over K=128, bias + sigmoid, store (16 x 6).
// ---------------------------------------------------------------------------
__global__ __launch_bounds__(WPB * 32) void sheaf_main_kernel(
    const float* __restrict__ xm, const float* __restrict__ em,
    const int* __restrict__ he, const float* __restrict__ gWg,
    const float* __restrict__ gBB, float* __restrict__ out) {
  __shared__ float sWg[16][PADW];
  __shared__ float sh[WPB][16][PADW];

  const int tid  = threadIdx.x;
  const int wave = tid >> 5;
  const int ln   = tid & 31;
  const int m16  = ln & 15;     // row of A / column of B this lane serves
  const int hf   = ln >> 4;     // half-wave: K offset {0,2} per f32 WMMA layout

  // stage Wg (gamma-folded, zero-padded W^T source) into padded LDS
  for (int i = tid; i < 16 * TWOF; i += WPB * 32) {
    sWg[i >> 7][i & 127] = gWg[i];
  }

  const int gtile = blockIdx.x * WPB + wave;           // exact: NNZ/16 tiles
  // indices for this tile; duplicated in upper half-wave (no divergence)
  const int rowi = he[(size_t)gtile * 16 + m16];
  const int coli = he[(size_t)NNZ_C + (size_t)gtile * 16 + m16];

  // ---- gather: 16 rows x (64 xm | 64 em) floats, float2 per lane ----
#pragma unroll
  for (int m = 0; m < 16; ++m) {
    int r = __builtin_amdgcn_readlane(rowi, m);
    int c = __builtin_amdgcn_readlane(coli, m);
    const float2* xr = (const float2*)(xm + (size_t)r * FF);
    const float2* er = (const float2*)(em + (size_t)c * FF);
    float2* hd = (float2*)(&sh[wave][m][0]);
    hd[ln]      = xr[ln];   // floats [0,64)
    hd[32 + ln] = er[ln];   // floats [64,128)
  }

  // ---- LayerNorm stats: lane covers half a row, combine halves via SWAPX16 ----
  float s1 = 0.0f, s2 = 0.0f;
  {
    const float2* sp = (const float2*)(&sh[wave][m16][hf * 64]);
#pragma unroll
    for (int i = 0; i < 32; ++i) {
      float2 v = sp[i];
      s1 += v.x + v.y;
      s2 += v.x * v.x + v.y * v.y;
    }
    // ds_swizzle group-of-32: xor=0x10, and=0x1f  -> partner lane ln^16
    float o1 = __int_as_float(__builtin_amdgcn_ds_swizzle(__float_as_int(s1), 0x401f));
    float o2 = __int_as_float(__builtin_amdgcn_ds_swizzle(__float_as_int(s2), 0x401f));
    s1 += o1;
    s2 += o2;
  }
  const float mu  = s1 * (1.0f / 128.0f);
  const float var = fmaxf(s2 * (1.0f / 128.0f) - mu * mu, 0.0f);
  const float rs  = rsqrtf(var + 1e-5f);
  const float mrs = -mu * rs;   // (h - mu)*rs == fma(h, rs, mrs)

  __syncthreads();   // Wg staging visible to all waves

  // ---- K=128 accumulation: 32 x wmma f32 16x16x4 ----
  v8f acc = {0.f, 0.f, 0.f, 0.f, 0.f, 0.f, 0.f, 0.f};
  const int koff = hf * 2;
  const float* hrow = &sh[wave][m16][0];
  const float* wrow = &sWg[m16][0];
#pragma unroll
  for (int c = 0; c < 32; ++c) {
    const int kb = c * 4 + koff;
    float2 hv = *(const float2*)(hrow + kb);
    float2 wv = *(const float2*)(wrow + kb);
    v2f a;  a.x = fmaf(hv.x, rs, mrs);  a.y = fmaf(hv.y, rs, mrs);
    v2f bm; bm.x = wv.x;                bm.y = wv.y;
    acc = __builtin_amdgcn_wmma_f32_16x16x4_f32(
        /*neg_a=*/false, a, /*neg_b=*/false, bm,
        /*c_mod=*/(short)0, acc, /*reuse_a=*/false, /*reuse_b=*/false);
  }

  // ---- epilogue: D[r][n], lane n = ln&15, rows r + 8*hf ----
  if (m16 < DD) {
    const float bias = gBB[m16];
    const int mbase = gtile * 16 + hf * 8;
#pragma unroll
    for (int r = 0; r < 8; ++r) {
      float v = acc[r] + bias;
      out[(size_t)(mbase + r) * DD + m16] = 1.0f / (1.0f + __expf(-v));
    }
  }
}

// ---------------------------------------------------------------------------
extern "C" void kernel_launch(void* const* d_in, const int* in_sizes, int n_in,
                              void* d_out, int out_size, void* d_ws, size_t ws_size,
                              hipStream_t stream) {
  (void)in_sizes; (void)n_in; (void)out_size; (void)ws_size;
  const float* x     = (const float*)d_in[0];
  const float* e     = (const float*)d_in[1];
  const int*   he    = (const int*)d_in[2];
  const float* gamma = (const float*)d_in[3];
  const float* beta  = (const float*)d_in[4];
  const float* W     = (const float*)d_in[5];
  const float* b     = (const float*)d_in[6];
  float* out = (float*)d_out;

  float* wsf = (float*)d_ws;
  float* xm  = wsf;                                  // 50000*64
  float* em  = xm + (size_t)NNODE * FF;              // 25000*64
  float* gWg = em + (size_t)NEDGE * FF;              // 16*128
  float* gBB = gWg + 16 * TWOF;                      // 16

  prep_w_kernel<<<1, 256, 0, stream>>>(W, gamma, beta, b, gWg, gBB);
  mean6_kernel<<<(NNODE * FF + 255) / 256, 256, 0, stream>>>(x, xm, NNODE * FF);
  mean6_kernel<<<(NEDGE * FF + 255) / 256, 256, 0, stream>>>(e, em, NEDGE * FF);

  // NNZ/16 = 100000 tiles, 4 tiles per 128-thread block -> 25000 blocks exact
  sheaf_main_kernel<<<NNZ_C / 16 / WPB, WPB * 32, 0, stream>>>(
      xm, em, he, gWg, gBB, out);
}